// GraphNNs_343597384356
// MI455X (gfx1250) — compile-verified
//
#include <hip/hip_runtime.h>

#define NN    50000      // nodes
#define NE    800000     // edges
#define DF    96         // feature dim
#define HIDD  256
#define NOUT  32
#define NG    64         // graphs
#define NL    5          // layers
#define EPSB  1e-5f

typedef __attribute__((ext_vector_type(2))) float v2f;
typedef __attribute__((ext_vector_type(8))) float v8f;

// ---------------- utility ----------------
__global__ void zero_f32(float* p, int n) {
    int i = blockIdx.x * blockDim.x + threadIdx.x;
    if (i < n) p[i] = 0.0f;
}

// ---------------- degree / edge weights ----------------
__global__ void deg_kernel(const int* __restrict__ ei, float* __restrict__ deg) {
    int e = blockIdx.x * blockDim.x + threadIdx.x;
    if (e < NE) atomicAdd(&deg[ei[e]], 1.0f);
}

__global__ void dinv_kernel(float* __restrict__ deg) {
    int i = blockIdx.x * blockDim.x + threadIdx.x;
    if (i < NN) {
        float d = deg[i];
        deg[i] = (d > 0.0f) ? rsqrtf(fmaxf(d, 1.0f)) : 0.0f;
    }
}

__global__ void ew_kernel(const int* __restrict__ ei, const float* __restrict__ dinv,
                          float* __restrict__ ew) {
    int e = blockIdx.x * blockDim.x + threadIdx.x;
    if (e < NE) ew[e] = -(dinv[ei[e]] * dinv[ei[NE + e]]);
}

// ---------------- edge scatter: Tx1[dst] += w * h[src]  (one wave per edge) ----------------
__global__ void scatter_kernel(const int* __restrict__ ei, const float* __restrict__ ew,
                               const float* __restrict__ h, float* __restrict__ tx1) {
    int wave = (int)((blockIdx.x * (size_t)blockDim.x + threadIdx.x) >> 5);
    int lane = threadIdx.x & 31;
    if (wave >= NE) return;
    int s = ei[wave];
    int d = ei[NE + wave];
    float w = ew[wave];
    const float* hs = h + (size_t)s * DF;
    float* td = tx1 + (size_t)d * DF;
#pragma unroll
    for (int f = lane; f < DF; f += 32) {
        atomicAdd(&td[f], w * hs[f]);
    }
}

// ---------------- WMMA f32 GEMM: Z = A@WA (+ B@WB) (+ bias), M,N mult of 16, K mult of 4 ----
// One wave computes one 16x16 output tile via V_WMMA_F32_16X16X4_F32.
__global__ void gemm_wmma_f32(const float* __restrict__ A, const float* __restrict__ WA,
                              const float* __restrict__ B, const float* __restrict__ WB,
                              const float* __restrict__ bias, float* __restrict__ Z,
                              int M, int K, int N) {
    int lane = threadIdx.x & 31;
    int wave = blockIdx.x * (blockDim.x >> 5) + (threadIdx.x >> 5);
    int ntiles = N >> 4;
    int total = (M >> 4) * ntiles;
    if (wave >= total) return;            // wave-uniform: EXEC stays all-1s for WMMA
    int tm = (wave / ntiles) << 4;
    int tn = (wave % ntiles) << 4;
    int hi = lane >> 4;                   // 0: lanes 0-15, 1: lanes 16-31
    int nl = lane & 15;
    int koff = hi << 1;                   // A/B K offset per ISA layout (K = 2*(l>>4)+r)

    v8f c = {};
    {
        const float* arow = A + (size_t)(tm + nl) * K;
        for (int k = 0; k < K; k += 4) {
            v2f a = *(const v2f*)(arow + k + koff);          // regs r=0,1 -> K=k+koff, k+koff+1
            v2f b;
            b.x = WA[(size_t)(k + koff) * N + tn + nl];
            b.y = WA[(size_t)(k + koff + 1) * N + tn + nl];
            c = __builtin_amdgcn_wmma_f32_16x16x4_f32(false, a, false, b, (short)0, c, false, false);
        }
    }
    if (B != nullptr) {
        const float* brow = B + (size_t)(tm + nl) * K;
        for (int k = 0; k < K; k += 4) {
            v2f a = *(const v2f*)(brow + k + koff);
            v2f b;
            b.x = WB[(size_t)(k + koff) * N + tn + nl];
            b.y = WB[(size_t)(k + koff + 1) * N + tn + nl];
            c = __builtin_amdgcn_wmma_f32_16x16x4_f32(false, a, false, b, (short)0, c, false, false);
        }
    }
    float bv = bias ? bias[tn + nl] : 0.0f;
#pragma unroll
    for (int v = 0; v < 8; ++v) {
        // C/D layout: reg v, lane l -> M = v + 8*(l>>4), N = l&15
        Z[(size_t)(tm + v + (hi << 3)) * N + tn + nl] = c[v] + bv;
    }
}

// ---------------- per-column mean/var (biased), one block per column ----------------
__global__ void colstats_kernel(const float* __restrict__ Z, int M, int N,
                                float* __restrict__ mean, float* __restrict__ var) {
    int ccol = blockIdx.x;
    float s = 0.0f, sq = 0.0f;
    for (int r = threadIdx.x; r < M; r += blockDim.x) {
        float v = Z[(size_t)r * N + ccol];
        s += v;
        sq += v * v;
    }
    __shared__ float ss[256];
    __shared__ float ssq[256];
    ss[threadIdx.x] = s;
    ssq[threadIdx.x] = sq;
    __syncthreads();
    for (int o = blockDim.x >> 1; o > 0; o >>= 1) {
        if ((int)threadIdx.x < o) {
            ss[threadIdx.x] += ss[threadIdx.x + o];
            ssq[threadIdx.x] += ssq[threadIdx.x + o];
        }
        __syncthreads();
    }
    if (threadIdx.x == 0) {
        float m = ss[0] / (float)M;
        mean[ccol] = m;
        var[ccol] = ssq[0] / (float)M - m * m;
    }
}

// ---------------- BN affine + ReLU ----------------
__global__ void bn_relu_kernel(const float* __restrict__ Z, const float* __restrict__ mean,
                               const float* __restrict__ var, const float* __restrict__ gamma,
                               const float* __restrict__ beta, float* __restrict__ H,
                               int M, int N) {
    int i = blockIdx.x * blockDim.x + threadIdx.x;
    if (i >= M * N) return;
    int cl = i % N;
    float v = (Z[i] - mean[cl]) * rsqrtf(var[cl] + EPSB) * gamma[cl] + beta[cl];
    H[i] = v > 0.0f ? v : 0.0f;
}

// ---------------- global mean pool ----------------
__global__ void pool_sum_kernel(const float* __restrict__ h, const int* __restrict__ batch,
                                float* __restrict__ gsum, float* __restrict__ gcnt) {
    int i = blockIdx.x * blockDim.x + threadIdx.x;
    if (i >= NN * DF) return;
    int node = i / DF;
    int f = i - node * DF;
    int b = batch[node];
    atomicAdd(&gsum[b * DF + f], h[i]);
    if (f == 0) atomicAdd(&gcnt[b], 1.0f);
}

__global__ void pool_div_kernel(const float* __restrict__ gsum, const float* __restrict__ gcnt,
                                float* __restrict__ g) {
    int i = blockIdx.x * blockDim.x + threadIdx.x;
    if (i < NG * DF) g[i] = gsum[i] / fmaxf(gcnt[i / DF], 1.0f);
}

// ---------------- host launcher ----------------
extern "C" void kernel_launch(void* const* d_in, const int* in_sizes, int n_in,
                              void* d_out, int out_size, void* d_ws, size_t ws_size,
                              hipStream_t stream) {
    const float* x        = (const float*)d_in[0];
    const int*   ei       = (const int*)d_in[1];
    const int*   batch    = (const int*)d_in[2];
    const float* W0       = (const float*)d_in[3];
    const float* W1       = (const float*)d_in[4];
    const float* cheb_b   = (const float*)d_in[5];
    const float* bn_gamma = (const float*)d_in[6];
    const float* bn_beta  = (const float*)d_in[7];
    const float* fc1_W    = (const float*)d_in[8];
    const float* fc1_b    = (const float*)d_in[9];
    const float* bnf_g    = (const float*)d_in[10];
    const float* bnf_b    = (const float*)d_in[11];
    const float* fc2_W    = (const float*)d_in[12];
    const float* fc2_b    = (const float*)d_in[13];
    float* out = (float*)d_out;

    // workspace carve (floats)
    float* ws   = (float*)d_ws;
    float* h    = ws;                      // NN*DF = 4,800,000
    float* z    = h + (size_t)NN * DF;     // 4,800,000
    float* tx1  = z + (size_t)NN * DF;     // 4,800,000
    float* deg  = tx1 + (size_t)NN * DF;   // 50,000 (becomes dinv in place)
    float* ew   = deg + NN;                // 800,000
    float* mean = ew + NE;                 // 256
    float* var  = mean + 256;              // 256
    float* gsum = var + 256;               // NG*DF
    float* gcnt = gsum + (size_t)NG * DF;  // NG
    float* g    = gcnt + NG;               // NG*DF
    float* f1   = g + (size_t)NG * DF;     // NG*HIDD

    const int T = 256;
    // degrees & edge weights (deg_w is layer-invariant)
    zero_f32<<<(NN + T - 1) / T, T, 0, stream>>>(deg, NN);
    deg_kernel<<<(NE + T - 1) / T, T, 0, stream>>>(ei, deg);
    dinv_kernel<<<(NN + T - 1) / T, T, 0, stream>>>(deg);
    ew_kernel<<<(NE + T - 1) / T, T, 0, stream>>>(ei, deg, ew);

    const int nzT = (NN * DF + T - 1) / T;            // 18750 blocks
    const int layerTiles = (NN / 16) * (DF / 16);     // 3125 * 6 = 18750 tiles
    const int gemmBlocks = (layerTiles + 3) / 4;      // 4 waves (tiles) per 128-thread block

    for (int l = 0; l < NL; ++l) {
        const float* in = (l == 0) ? x : h;
        // Tx1 = segment_sum over dst of deg_w * in[src]
        zero_f32<<<nzT, T, 0, stream>>>(tx1, NN * DF);
        scatter_kernel<<<(NE * 32 + T - 1) / T, T, 0, stream>>>(ei, ew, in, tx1);
        // z = in@W0 + Tx1@W1 + b   (WMMA f32)
        gemm_wmma_f32<<<gemmBlocks, 128, 0, stream>>>(in, W0, tx1, W1, cheb_b, z, NN, DF, DF);
        // batchnorm (training-mode stats) + relu
        colstats_kernel<<<DF, 256, 0, stream>>>(z, NN, DF, mean, var);
        bn_relu_kernel<<<nzT, T, 0, stream>>>(z, mean, var, bn_gamma + l * DF, bn_beta + l * DF,
                                              h, NN, DF);
    }

    // global mean pool
    zero_f32<<<(NG * DF + NG + T - 1) / T, T, 0, stream>>>(gsum, NG * DF + NG); // gsum + gcnt
    pool_sum_kernel<<<nzT, T, 0, stream>>>(h, batch, gsum, gcnt);
    pool_div_kernel<<<(NG * DF + T - 1) / T, T, 0, stream>>>(gsum, gcnt, g);

    // fc1: [64x96]@[96x256] + b -> BN -> relu
    {
        int tiles = (NG / 16) * (HIDD / 16);          // 4*16 = 64
        gemm_wmma_f32<<<(tiles + 3) / 4, 128, 0, stream>>>(g, fc1_W, nullptr, nullptr, fc1_b,
                                                           f1, NG, DF, HIDD);
        colstats_kernel<<<HIDD, 64, 0, stream>>>(f1, NG, HIDD, mean, var);
        bn_relu_kernel<<<(NG * HIDD + T - 1) / T, T, 0, stream>>>(f1, mean, var, bnf_g, bnf_b,
                                                                  f1, NG, HIDD);
    }
    // fc2: [64x256]@[256x32] + b -> output
    {
        int tiles = (NG / 16) * (NOUT / 16);          // 4*2 = 8
        gemm_wmma_f32<<<(tiles + 3) / 4, 128, 0, stream>>>(f1, fc2_W, nullptr, nullptr, fc2_b,
                                                           out, NG, HIDD, NOUT);
    }

    (void)in_sizes; (void)n_in; (void)out_size; (void)ws_size;
}